// Cifar10_JointMembership_89816356094197
// MI455X (gfx1250) — compile-verified
//
#include <hip/hip_runtime.h>

// Closed form of the 2-qubit circuit (see analysis):
//   out[b,p,0,c] = 0.5*(cos(theta_c)*cos(a) - sin(theta_c)*sin(a)*sin(b)) + 0.5
//   out[b,p,1,c] = 0.5*(cos(a)*cos(b)) + 0.5            (class independent)
// Cast as a 16x16x4 f32 WMMA per wave: 8 pairs -> 16 rows (even=z0, odd=z1),
// K=4 features, B holds per-class rotation + bias columns. Store-BW bound
// (~54 MB traffic -> ~2.3 us floor @ 23.3 TB/s).

typedef __attribute__((ext_vector_type(2))) float v2f;
typedef __attribute__((ext_vector_type(8))) float v8f;

#define PIXELS    3072
#define N_PAIRS   460
#define N_CLASSES 10
#define BATCH     1024

__global__ __launch_bounds__(256) void qpair_wmma_kernel(
    const float* __restrict__ x,        // [BATCH, 3072]
    const float* __restrict__ theta,    // [10]
    const int*   __restrict__ pair_idx, // [BATCH, N_PAIRS, 2]
    float* __restrict__ out)            // [BATCH, N_PAIRS*2, 10]
{
    const int lane      = threadIdx.x & 31;
    const int waveInBlk = threadIdx.x >> 5;
    const int wave      = blockIdx.x * (blockDim.x >> 5) + waveInBlk;
    const int P0        = wave * 8;        // first pair of this wave's 8-pair tile

    const int half   = lane >> 4;          // 0: lanes 0-15, 1: lanes 16-31
    const int m      = lane & 15;          // A row / B,D column index
    const int p      = m >> 1;             // pair within tile (0..7)
    const int oddRow = m & 1;              // 0 = z0 row, 1 = z1 row

    // --- gather the pair's two pixel angles (x4 lane redundancy, cache-hit) ---
    const int gp = P0 + p;                 // global pair index (< 471040)
    const int b  = gp / N_PAIRS;
    const int ia = pair_idx[gp * 2 + 0];
    const int ib = pair_idx[gp * 2 + 1];
    const float va = x[b * PIXELS + ia];
    const float vb = x[b * PIXELS + ib];

    const float ca = __cosf(va), sa = __sinf(va);
    const float cb = __cosf(vb), sb = __sinf(vb);
    const float f0 = ca;           // K0 feature (even rows)
    const float f1 = sa * sb;      // K1 feature (even rows)
    const float f2 = ca * cb;      // K2 feature (odd rows)

    // A 16x4 f32 layout: lanes 0-15 hold K0(v0),K1(v1); lanes 16-31 hold K2(v0),K3(v1).
    // Branch-free selects keep EXEC all-ones for the WMMA.
    v2f A;
    A.x = half ? (oddRow ? f2 : 0.0f) : (oddRow ? 0.0f : f0);
    A.y = half ? 1.0f                 : (oddRow ? 0.0f : f1);

    // B 4x16 f32 layout (half-offset row striping like C/D):
    //   v0: lanes 0-15 = row K0, lanes 16-31 = row K2
    //   v1: lanes 0-15 = row K1, lanes 16-31 = row K3
    const int   cIdx = (m < N_CLASSES) ? m : (N_CLASSES - 1); // clamp: no OOB, full EXEC
    const float th   = theta[cIdx];
    const float ok   = (m < N_CLASSES) ? 1.0f : 0.0f;
    v2f Bm;
    Bm.x = half ? 0.5f : ( 0.5f * __cosf(th) * ok);   // K2 row : K0 row
    Bm.y = half ? 0.5f : (-0.5f * __sinf(th) * ok);   // K3 row : K1 row

    v8f C = {};
    // 8 args: (neg_a, A, neg_b, B, c_mod, C, reuse_a, reuse_b)
    v8f D = __builtin_amdgcn_wmma_f32_16x16x4_f32(
        false, A, false, Bm, (short)0, C, false, false);

    // D layout: VGPR r holds row M=r (lanes 0-15) and M=8+r (lanes 16-31), col = m.
    // Output row (P0*2 + M) has 10 contiguous floats; lanes 0-9 / 16-25 coalesce.
    if (m < N_CLASSES) {
        const int rowBase = (P0 * 2 + half * 8) * N_CLASSES + m;
        #pragma unroll
        for (int r = 0; r < 8; ++r) {
            out[rowBase + r * N_CLASSES] = D[r];
        }
    }
}

extern "C" void kernel_launch(void* const* d_in, const int* in_sizes, int n_in,
                              void* d_out, int out_size, void* d_ws, size_t ws_size,
                              hipStream_t stream) {
    const float* x        = (const float*)d_in[0];
    const float* theta    = (const float*)d_in[1];
    const int*   pair_idx = (const int*)d_in[2];
    float*       out      = (float*)d_out;

    // 471,040 pairs total; 8 waves/block * 8 pairs/wave = 64 pairs/block -> 7,360 blocks, no tail.
    const int totalPairs    = BATCH * N_PAIRS;
    const int pairsPerBlock = 64;
    dim3 grid(totalPairs / pairsPerBlock);
    dim3 block(256);
    qpair_wmma_kernel<<<grid, block, 0, stream>>>(x, theta, pair_idx, out);
}